// ProbabilityEncoder_4535485464988
// MI455X (gfx1250) — compile-verified
//
#include <hip/hip_runtime.h>

// ---------------------------------------------------------------------------
// ProbabilityEncoder for MI455X (gfx1250): MLP encoder + t-student pairwise
// probabilities. N=8192, DIM=128, widths 64/32/16, EMB=16.
// Strategy: fp32 WMMA (V_WMMA_F32_16X16X4_F32) everywhere; recompute the
// tiny Gram (z z^T, K=16, z = 512KB -> L2 resident) twice so the 256MB
// output is written exactly once (HBM-bound: ~11us at 23.3 TB/s).
// ---------------------------------------------------------------------------

typedef float v2f __attribute__((ext_vector_type(2)));
typedef float v8f __attribute__((ext_vector_type(8)));

#define N_PTS 8192
#define NT    512     // 16-row tiles

static __device__ __forceinline__ v8f wmma_f32_4(v2f a, v2f b, v8f c) {
  // D = A(16x4) * B(4x16) + C ; f32 in/out
  return __builtin_amdgcn_wmma_f32_16x16x4_f32(false, a, false, b, (short)0, c,
                                               false, false);
}

// ---------------------------------------------------------------------------
// Kernel 1: encoder. One wave per 16-row tile of x. 8 waves / block.
// Layer outputs staged in LDS with padded strides (68/36/20 words) so the
// two half-wave rows hit disjoint LDS banks.
// ---------------------------------------------------------------------------
__global__ __launch_bounds__(256) void encoder_kernel(
    const float* __restrict__ x,
    const float* __restrict__ W1, const float* __restrict__ b1,
    const float* __restrict__ W2, const float* __restrict__ b2,
    const float* __restrict__ W3, const float* __restrict__ b3,
    const float* __restrict__ W4, const float* __restrict__ b4,
    float* __restrict__ z, float* __restrict__ sq)
{
  __shared__ float ldsA[8][16 * 68];   // 16x64 @ stride 68 (also 16x16 @ 20)
  __shared__ float ldsB[8][16 * 36];   // 16x32 @ stride 36 (also 16x16 @ 20)

  const int wave = threadIdx.x >> 5;
  const int lane = threadIdx.x & 31;
  const int half = lane >> 4;          // 0: K{0,1} / rows 0-7 ; 1: K{2,3} / rows 8-15
  const int lr   = lane & 15;
  const int tile = blockIdx.x * 8 + wave;      // 0..511
  const int row0 = tile * 16;

  float* bA = ldsA[wave];
  float* bB = ldsB[wave];

  // ---- Layer 1: X(16x128) @ W1(128x64) -> bA (stride 68) --------------------
  {
    v8f acc0 = 0.f, acc1 = 0.f, acc2 = 0.f, acc3 = 0.f;
    const float* xrow = x + (size_t)(row0 + lr) * 128;
    #pragma unroll 4
    for (int kb = 0; kb < 32; ++kb) {
      const int k0 = kb * 4 + 2 * half;
      v2f a; a.x = xrow[k0]; a.y = xrow[k0 + 1];
      v2f w;
      w.x = W1[(k0    ) * 64 +  0 + lr]; w.y = W1[(k0 + 1) * 64 +  0 + lr];
      acc0 = wmma_f32_4(a, w, acc0);
      w.x = W1[(k0    ) * 64 + 16 + lr]; w.y = W1[(k0 + 1) * 64 + 16 + lr];
      acc1 = wmma_f32_4(a, w, acc1);
      w.x = W1[(k0    ) * 64 + 32 + lr]; w.y = W1[(k0 + 1) * 64 + 32 + lr];
      acc2 = wmma_f32_4(a, w, acc2);
      w.x = W1[(k0    ) * 64 + 48 + lr]; w.y = W1[(k0 + 1) * 64 + 48 + lr];
      acc3 = wmma_f32_4(a, w, acc3);
    }
    v8f acc[4] = {acc0, acc1, acc2, acc3};
    #pragma unroll
    for (int n = 0; n < 4; ++n) {
      const float bias = b1[n * 16 + lr];
      #pragma unroll
      for (int r = 0; r < 8; ++r) {
        float v = acc[n][r] + bias;
        bA[(r + 8 * half) * 68 + n * 16 + lr] = v > 0.f ? v : 0.f;
      }
    }
  }

  // ---- Layer 2: H1(16x64) @ W2(64x32) -> bB (stride 36) ---------------------
  {
    v8f acc0 = 0.f, acc1 = 0.f;
    #pragma unroll
    for (int kb = 0; kb < 16; ++kb) {
      const int k0 = kb * 4 + 2 * half;
      v2f a; a.x = bA[lr * 68 + k0]; a.y = bA[lr * 68 + k0 + 1];
      v2f w;
      w.x = W2[(k0    ) * 32 +  0 + lr]; w.y = W2[(k0 + 1) * 32 +  0 + lr];
      acc0 = wmma_f32_4(a, w, acc0);
      w.x = W2[(k0    ) * 32 + 16 + lr]; w.y = W2[(k0 + 1) * 32 + 16 + lr];
      acc1 = wmma_f32_4(a, w, acc1);
    }
    v8f acc[2] = {acc0, acc1};
    #pragma unroll
    for (int n = 0; n < 2; ++n) {
      const float bias = b2[n * 16 + lr];
      #pragma unroll
      for (int r = 0; r < 8; ++r) {
        float v = acc[n][r] + bias;
        bB[(r + 8 * half) * 36 + n * 16 + lr] = v > 0.f ? v : 0.f;
      }
    }
  }

  // ---- Layer 3: H2(16x32) @ W3(32x16) -> bA (stride 20) ---------------------
  {
    v8f acc0 = 0.f;
    #pragma unroll
    for (int kb = 0; kb < 8; ++kb) {
      const int k0 = kb * 4 + 2 * half;
      v2f a; a.x = bB[lr * 36 + k0]; a.y = bB[lr * 36 + k0 + 1];
      v2f w; w.x = W3[k0 * 16 + lr]; w.y = W3[(k0 + 1) * 16 + lr];
      acc0 = wmma_f32_4(a, w, acc0);
    }
    const float bias = b3[lr];
    #pragma unroll
    for (int r = 0; r < 8; ++r) {
      float v = acc0[r] + bias;
      bA[(r + 8 * half) * 20 + lr] = v > 0.f ? v : 0.f;
    }
  }

  // ---- Layer 4: H3(16x16) @ W4(16x16) -> z, sq ------------------------------
  {
    v8f acc0 = 0.f;
    #pragma unroll
    for (int kb = 0; kb < 4; ++kb) {
      const int k0 = kb * 4 + 2 * half;
      v2f a; a.x = bA[lr * 20 + k0]; a.y = bA[lr * 20 + k0 + 1];
      v2f w; w.x = W4[k0 * 16 + lr]; w.y = W4[(k0 + 1) * 16 + lr];
      acc0 = wmma_f32_4(a, w, acc0);
    }
    const float bias = b4[lr];
    #pragma unroll
    for (int r = 0; r < 8; ++r) {
      float v = acc0[r] + bias;
      z[(size_t)(row0 + r + 8 * half) * 16 + lr] = v;
      bB[(r + 8 * half) * 20 + lr] = v;      // stage for norm computation
    }
    if (half == 0) {                          // lanes 0..15: one row each
      float s = 0.f;
      #pragma unroll
      for (int c = 0; c < 16; ++c) { float t = bB[lr * 20 + c]; s += t * t; }
      sq[row0 + lr] = s;
    }
  }
}

// ---------------------------------------------------------------------------
// Kernel 2: zero the row-sum accumulator.
// ---------------------------------------------------------------------------
__global__ void zero_kernel(float* __restrict__ S) {
  const int i = blockIdx.x * 256 + threadIdx.x;
  if (i < N_PTS) S[i] = 0.f;
}

// ---------------------------------------------------------------------------
// Kernel 3: row sums  S[i] = sum_j 1/(1+dist(i,j)).
// grid = (8, 512): blockIdx.y = i-tile; 64 waves stripe the 512 j-tiles.
// Gram tile via 4x V_WMMA_F32_16X16X4_F32; 16-lane shuffle reduce; f32 atomic.
// ---------------------------------------------------------------------------
__global__ __launch_bounds__(256) void rowsum_kernel(
    const float* __restrict__ z, const float* __restrict__ sq,
    float* __restrict__ S)
{
  const int wave = threadIdx.x >> 5;
  const int lane = threadIdx.x & 31;
  const int half = lane >> 4;
  const int lr   = lane & 15;
  const int i0   = blockIdx.y * 16;
  const int jw   = blockIdx.x * 8 + wave;    // 0..63

  // A-fragment source: z row (i0+lr), all K=16
  float za[16];
  {
    const float* zi = z + (size_t)(i0 + lr) * 16;
    #pragma unroll
    for (int k = 0; k < 16; ++k) za[k] = zi[k];
  }
  float sqi[8];
  #pragma unroll
  for (int r = 0; r < 8; ++r) sqi[r] = sq[i0 + r + 8 * half];

  float rsum[8];
  #pragma unroll
  for (int r = 0; r < 8; ++r) rsum[r] = 0.f;

  for (int jt = jw; jt < NT; jt += 64) {
    const int j0 = jt * 16;
    const float* zj = z + (size_t)(j0 + lr) * 16;   // B: column n = lr
    float zb[16];
    #pragma unroll
    for (int k = 0; k < 16; ++k) zb[k] = zj[k];
    const float sqj = sq[j0 + lr];

    v8f acc = 0.f;
    #pragma unroll
    for (int kb = 0; kb < 4; ++kb) {
      const int k0 = kb * 4 + 2 * half;
      v2f a; a.x = za[k0]; a.y = za[k0 + 1];
      v2f b; b.x = zb[k0]; b.y = zb[k0 + 1];
      acc = wmma_f32_4(a, b, acc);
    }
    #pragma unroll
    for (int r = 0; r < 8; ++r) {
      float d = sqi[r] + sqj - 2.f * acc[r];
      d = d > 0.f ? d : 0.f;
      rsum[r] += __builtin_amdgcn_rcpf(1.f + d);
    }
  }

  // reduce the 16 columns (lanes within each half), then one atomic per row
  #pragma unroll
  for (int r = 0; r < 8; ++r) {
    float s = rsum[r];
    #pragma unroll
    for (int m = 1; m < 16; m <<= 1) s += __shfl_xor(s, m, 32);
    if (lr == 0) atomicAdd(&S[i0 + r + 8 * half], s);
  }
}

// ---------------------------------------------------------------------------
// Kernel 4: S -> 1/S (exact divide, only 8192 elements).
// ---------------------------------------------------------------------------
__global__ void invert_kernel(float* __restrict__ S) {
  const int i = blockIdx.x * 256 + threadIdx.x;
  if (i < N_PTS) S[i] = 1.f / S[i];
}

// ---------------------------------------------------------------------------
// Kernel 5: P[i,j] = (1/(1+dist)) * invS[i]. One wave per 16x16 tile.
// The single pass that touches the 256MB output.
// ---------------------------------------------------------------------------
__global__ __launch_bounds__(256) void prob_kernel(
    const float* __restrict__ z, const float* __restrict__ sq,
    const float* __restrict__ invS, float* __restrict__ out)
{
  const int wave = threadIdx.x >> 5;
  const int lane = threadIdx.x & 31;
  const int half = lane >> 4;
  const int lr   = lane & 15;
  const int i0   = blockIdx.y * 16;
  const int j0   = (blockIdx.x * 8 + wave) * 16;

  float za[16], zb[16];
  {
    const float* zi = z + (size_t)(i0 + lr) * 16;
    const float* zj = z + (size_t)(j0 + lr) * 16;
    #pragma unroll
    for (int k = 0; k < 16; ++k) { za[k] = zi[k]; zb[k] = zj[k]; }
  }
  float sqi[8], isi[8];
  #pragma unroll
  for (int r = 0; r < 8; ++r) {
    sqi[r] = sq[i0 + r + 8 * half];
    isi[r] = invS[i0 + r + 8 * half];
  }
  const float sqj = sq[j0 + lr];

  v8f acc = 0.f;
  #pragma unroll
  for (int kb = 0; kb < 4; ++kb) {
    const int k0 = kb * 4 + 2 * half;
    v2f a; a.x = za[k0]; a.y = za[k0 + 1];
    v2f b; b.x = zb[k0]; b.y = zb[k0 + 1];
    acc = wmma_f32_4(a, b, acc);
  }

  #pragma unroll
  for (int r = 0; r < 8; ++r) {
    float d = sqi[r] + sqj - 2.f * acc[r];
    d = d > 0.f ? d : 0.f;
    const float p = __builtin_amdgcn_rcpf(1.f + d) * isi[r];
    out[(size_t)(i0 + r + 8 * half) * N_PTS + j0 + lr] = p;
  }
}

// ---------------------------------------------------------------------------
extern "C" void kernel_launch(void* const* d_in, const int* in_sizes, int n_in,
                              void* d_out, int out_size, void* d_ws, size_t ws_size,
                              hipStream_t stream) {
  const float* x  = (const float*)d_in[0];
  const float* W1 = (const float*)d_in[1];
  const float* b1 = (const float*)d_in[2];
  const float* W2 = (const float*)d_in[3];
  const float* b2 = (const float*)d_in[4];
  const float* W3 = (const float*)d_in[5];
  const float* b3 = (const float*)d_in[6];
  const float* W4 = (const float*)d_in[7];
  const float* b4 = (const float*)d_in[8];
  float* out = (float*)d_out;

  float* z  = (float*)d_ws;          // 8192*16
  float* sq = z + (size_t)N_PTS * 16; // 8192
  float* S  = sq + N_PTS;             // 8192

  encoder_kernel<<<dim3(NT / 8), 256, 0, stream>>>(x, W1, b1, W2, b2, W3, b3,
                                                   W4, b4, z, sq);
  zero_kernel<<<dim3(N_PTS / 256), 256, 0, stream>>>(S);
  rowsum_kernel<<<dim3(8, NT), 256, 0, stream>>>(z, sq, S);
  invert_kernel<<<dim3(N_PTS / 256), 256, 0, stream>>>(S);
  prob_kernel<<<dim3(NT / 8, NT), 256, 0, stream>>>(z, sq, S, out);
}